// DecoderLSTM_33028298506681
// MI455X (gfx1250) — compile-verified
//
#include <hip/hip_runtime.h>
#include <math.h>

// ---------------------------------------------------------------------------
// Problem constants (DecoderLSTM): H=1024, L=2, B=64, S=1024
// ---------------------------------------------------------------------------
#define Hh 1024
#define Bb 64
#define Ss 1024
#define BH (Bb * Hh)          // 65536
#define NCHUNK 4              // S-chunks for online-softmax partials
#define T4 (BH / 4)           // 16384 float4 positions

typedef float v2f __attribute__((ext_vector_type(2)));
typedef float v8f __attribute__((ext_vector_type(8)));

__device__ __forceinline__ float sigmoid_(float x) {
    return 1.0f / (1.0f + __expf(-x));
}
__device__ __forceinline__ float tanh_(float x) {
    float e = __expf(2.0f * x);
    return 1.0f - 2.0f / (e + 1.0f);   // correct limits at +-inf
}

// ---------------------------------------------------------------------------
// Kernel 1: attn_lin[b,n] = sum_k x[b,k] * attn_W[n,k] + attn_b[n]
// M=64, N=1024, K=1024.  8 waves/block: wave w -> M-tile (w&3), N-subtile (w>>2).
// grid.x = N/32.
// ---------------------------------------------------------------------------
__global__ __launch_bounds__(256) void attn_gemm_kernel(
    const float* __restrict__ X, const float* __restrict__ W,
    const float* __restrict__ bias, float* __restrict__ out, int K, int N)
{
    const int tid  = threadIdx.x;
    const int w    = tid >> 5;
    const int lane = tid & 31;
    const int half = lane >> 4;
    const int ln   = lane & 15;
    const int m0   = (w & 3) * 16;
    const int n    = blockIdx.x * 32 + (w >> 2) * 16 + ln;

    const float bv = bias[n];
    v8f acc;
#pragma unroll
    for (int r = 0; r < 8; ++r) acc[r] = bv;

    const float* xrow = X + (size_t)(m0 + ln) * K;   // A: lane owns row M = m0+ln
    const float* wrow = W + (size_t)n * K;           // B: lane owns col N = n (row of W)
    const int kb = 2 * half;                         // K pair per lane half

    for (int k0 = 0; k0 < K; k0 += 4) {
        v2f a = *(const v2f*)(xrow + k0 + kb);
        v2f b = *(const v2f*)(wrow + k0 + kb);
        acc = __builtin_amdgcn_wmma_f32_16x16x4_f32(
            false, a, false, b, (short)0, acc, false, false);
    }
#pragma unroll
    for (int r = 0; r < 8; ++r)
        out[(size_t)(m0 + r + 8 * half) * N + n] = acc[r];
}

// ---------------------------------------------------------------------------
// Kernel 2: online-softmax partials over an S-chunk.
// Per (b,h): score(s) = attn_lin[b,h] * enc[s,b,h]; keep (m, l=sum e, acc=sum e*v).
// Thread owns 4 consecutive h (float4). grid = (T4/256, NCHUNK).
// part layout: float4 triples [chunk*T4 + t][3] = {m, l, acc}
// ---------------------------------------------------------------------------
__global__ __launch_bounds__(256) void attn_softmax_partial_kernel(
    const float* __restrict__ attn_lin, const float* __restrict__ enc,
    float4* __restrict__ part)
{
    const int t     = blockIdx.x * 256 + threadIdx.x;   // 0 .. T4-1
    const int chunk = blockIdx.y;
    const float4 a  = ((const float4*)attn_lin)[t];
    const float4* enc4 = (const float4*)enc;

    float4 m   = make_float4(-INFINITY, -INFINITY, -INFINITY, -INFINITY);
    float4 l   = make_float4(0.f, 0.f, 0.f, 0.f);
    float4 acc = make_float4(0.f, 0.f, 0.f, 0.f);

    const int s0 = chunk * (Ss / NCHUNK);
    const int s1 = s0 + (Ss / NCHUNK);
    for (int s = s0; s < s1; ++s) {
        float4 v = enc4[(size_t)s * T4 + t];
        if (s + 8 < s1)   // pull next rows toward L2 (global_prefetch)
            __builtin_prefetch(&enc4[(size_t)(s + 8) * T4 + t], 0, 0);
#define ONLSM(C)                                            \
        {                                                   \
            float sc = a.C * v.C;                           \
            float mn = fmaxf(m.C, sc);                      \
            float sca = __expf(m.C - mn);                   \
            float p   = __expf(sc - mn);                    \
            l.C   = l.C * sca + p;                          \
            acc.C = acc.C * sca + p * v.C;                  \
            m.C   = mn;                                     \
        }
        ONLSM(x) ONLSM(y) ONLSM(z) ONLSM(w)
#undef ONLSM
    }
    const size_t base = ((size_t)chunk * T4 + t) * 3;
    part[base + 0] = m;
    part[base + 1] = l;
    part[base + 2] = acc;
}

// ---------------------------------------------------------------------------
// Kernel 3: merge NCHUNK partials -> attention[b,h] = acc / l
// ---------------------------------------------------------------------------
__global__ __launch_bounds__(256) void attn_softmax_combine_kernel(
    const float4* __restrict__ part, float* __restrict__ attention)
{
    const int t = blockIdx.x * 256 + threadIdx.x;   // 0 .. T4-1
    float4 m   = part[(size_t)t * 3 + 0];
    float4 l   = part[(size_t)t * 3 + 1];
    float4 acc = part[(size_t)t * 3 + 2];
    for (int c = 1; c < NCHUNK; ++c) {
        const size_t base = ((size_t)c * T4 + t) * 3;
        float4 m2   = part[base + 0];
        float4 l2   = part[base + 1];
        float4 acc2 = part[base + 2];
#define MERGE(C)                                            \
        {                                                   \
            float mn = fmaxf(m.C, m2.C);                    \
            float s1 = __expf(m.C - mn);                    \
            float s2 = __expf(m2.C - mn);                   \
            l.C   = l.C * s1 + l2.C * s2;                   \
            acc.C = acc.C * s1 + acc2.C * s2;               \
            m.C   = mn;                                     \
        }
        MERGE(x) MERGE(y) MERGE(z) MERGE(w)
#undef MERGE
    }
    float4 o;
    o.x = acc.x / l.x; o.y = acc.y / l.y; o.z = acc.z / l.z; o.w = acc.w / l.w;
    ((float4*)attention)[t] = o;
}

// ---------------------------------------------------------------------------
// Kernels 4/5: fused LSTM layer.
//   gates[b, gi*H+n] = actA@Wih[:, :KA].T + actB@Wih[:, KA:].T + h@Whh.T + b_ih + b_hh
//   then cell nonlinearity through LDS.
// Block: 512 thr = 16 waves = 4 gates x 4 M-tiles; 16 output columns per block.
// grid.x = H/16 = 64.
// ---------------------------------------------------------------------------
__global__ __launch_bounds__(512) void lstm_layer_kernel(
    const float* __restrict__ actA, int KA,
    const float* __restrict__ actB, int KB,        // nullable (KB==0)
    const float* __restrict__ hprev,               // B x H
    const float* __restrict__ W_ih,                // 4H x (KA+KB)
    const float* __restrict__ W_hh,                // 4H x H
    const float* __restrict__ b_ih, const float* __restrict__ b_hh,
    const float* __restrict__ c_in,                // B x H
    float* __restrict__ h_out, float* __restrict__ h_out2,  // h_out2 nullable
    float* __restrict__ c_out)
{
    __shared__ float lds_g[4][Bb][16];             // 16 KB gate tiles

    const int tid  = threadIdx.x;
    const int w    = tid >> 5;
    const int lane = tid & 31;
    const int half = lane >> 4;
    const int ln   = lane & 15;
    const int gi   = w >> 2;                       // gate: 0=i 1=f 2=g 3=o
    const int m0   = (w & 3) * 16;
    const int n0   = blockIdx.x * 16;
    const int nG   = gi * Hh + n0 + ln;            // row of W_ih / W_hh
    const int Kih  = KA + KB;
    const int kb   = 2 * half;

    const float bv = b_ih[nG] + b_hh[nG];
    v8f acc;
#pragma unroll
    for (int r = 0; r < 8; ++r) acc[r] = bv;

    // --- actA @ W_ih[:, 0:KA].T ---
    {
        const float* xrow = actA + (size_t)(m0 + ln) * KA;
        const float* wrow = W_ih + (size_t)nG * Kih;
        for (int k0 = 0; k0 < KA; k0 += 4) {
            v2f a = *(const v2f*)(xrow + k0 + kb);
            v2f b = *(const v2f*)(wrow + k0 + kb);
            acc = __builtin_amdgcn_wmma_f32_16x16x4_f32(
                false, a, false, b, (short)0, acc, false, false);
        }
    }
    // --- actB @ W_ih[:, KA:KA+KB].T ---
    if (KB > 0) {
        const float* xrow = actB + (size_t)(m0 + ln) * KB;
        const float* wrow = W_ih + (size_t)nG * Kih + KA;
        for (int k0 = 0; k0 < KB; k0 += 4) {
            v2f a = *(const v2f*)(xrow + k0 + kb);
            v2f b = *(const v2f*)(wrow + k0 + kb);
            acc = __builtin_amdgcn_wmma_f32_16x16x4_f32(
                false, a, false, b, (short)0, acc, false, false);
        }
    }
    // --- hprev @ W_hh.T ---
    {
        const float* xrow = hprev + (size_t)(m0 + ln) * Hh;
        const float* wrow = W_hh + (size_t)nG * Hh;
        for (int k0 = 0; k0 < Hh; k0 += 4) {
            v2f a = *(const v2f*)(xrow + k0 + kb);
            v2f b = *(const v2f*)(wrow + k0 + kb);
            acc = __builtin_amdgcn_wmma_f32_16x16x4_f32(
                false, a, false, b, (short)0, acc, false, false);
        }
    }

    // stash gate tile: C/D layout -> row = m0 + r + 8*half, col = ln
#pragma unroll
    for (int r = 0; r < 8; ++r)
        lds_g[gi][m0 + r + 8 * half][ln] = acc[r];

    __syncthreads();

    // cell nonlinearity: 64x16 elements, 512 threads -> 2 each
#pragma unroll
    for (int j = 0; j < 2; ++j) {
        const int e  = tid + j * 512;
        const int b  = e >> 4;
        const int nn = e & 15;
        const int n  = n0 + nn;
        const float iv = sigmoid_(lds_g[0][b][nn]);
        const float fv = sigmoid_(lds_g[1][b][nn]);
        const float gv = tanh_(lds_g[2][b][nn]);
        const float ov = sigmoid_(lds_g[3][b][nn]);
        const float cn = fv * c_in[(size_t)b * Hh + n] + iv * gv;
        const float hn = ov * tanh_(cn);
        c_out[(size_t)b * Hh + n] = cn;
        h_out[(size_t)b * Hh + n] = hn;
        if (h_out2) h_out2[(size_t)b * Hh + n] = hn;
    }
}

// ---------------------------------------------------------------------------
// Launch
// ---------------------------------------------------------------------------
extern "C" void kernel_launch(void* const* d_in, const int* in_sizes, int n_in,
                              void* d_out, int out_size, void* d_ws, size_t ws_size,
                              hipStream_t stream)
{
    (void)in_sizes; (void)n_in; (void)out_size; (void)ws_size;

    const float* x      = (const float*)d_in[0];   // (1,B,H)
    const float* h0     = (const float*)d_in[1];   // (L,B,H)
    const float* c0     = (const float*)d_in[2];   // (L,B,H)
    const float* enc    = (const float*)d_in[3];   // (S,B,H)
    const float* attn_W = (const float*)d_in[4];   // (H,H)
    const float* attn_b = (const float*)d_in[5];   // (H)
    const float* W_ih0  = (const float*)d_in[6];   // (4H,2H)
    const float* W_hh0  = (const float*)d_in[7];   // (4H,H)
    const float* b_ih0  = (const float*)d_in[8];
    const float* b_hh0  = (const float*)d_in[9];
    const float* W_ih1  = (const float*)d_in[10];  // (4H,H)
    const float* W_hh1  = (const float*)d_in[11];  // (4H,H)
    const float* b_ih1  = (const float*)d_in[12];
    const float* b_hh1  = (const float*)d_in[13];

    float* out = (float*)d_out;
    // d_out layout: output(BH)=h1 | h_n: h0_new,h1_new | c_n: c0_new,c1_new
    float* out_h1a = out;                 // output
    float* out_h0  = out + (size_t)BH;    // h_n[0]
    float* out_h1b = out + 2 * (size_t)BH;// h_n[1]
    float* out_c0  = out + 3 * (size_t)BH;// c_n[0]
    float* out_c1  = out + 4 * (size_t)BH;// c_n[1]

    float* ws        = (float*)d_ws;
    float* ws_lin    = ws;                        // attn_lin: BH
    float* ws_att    = ws + (size_t)BH;           // attention: BH
    float4* ws_part  = (float4*)(ws + 2 * (size_t)BH); // NCHUNK*T4*3 float4

    // 1) attn_lin = x @ attn_W.T + attn_b
    attn_gemm_kernel<<<Hh / 32, 256, 0, stream>>>(x, attn_W, attn_b, ws_lin, Hh, Hh);

    // 2) online-softmax partials over enc_seq (single streaming pass)
    attn_softmax_partial_kernel<<<dim3(T4 / 256, NCHUNK), 256, 0, stream>>>(
        ws_lin, enc, ws_part);

    // 3) merge partials -> attention
    attn_softmax_combine_kernel<<<T4 / 256, 256, 0, stream>>>(ws_part, ws_att);

    // 4) layer 0: lstm_in = [x, attention]
    lstm_layer_kernel<<<Hh / 16, 512, 0, stream>>>(
        x, Hh, ws_att, Hh, h0, W_ih0, W_hh0, b_ih0, b_hh0,
        c0, out_h0, nullptr, out_c0);

    // 5) layer 1: input = h0_new
    lstm_layer_kernel<<<Hh / 16, 512, 0, stream>>>(
        out_h0, Hh, nullptr, 0, h0 + BH, W_ih1, W_hh1, b_ih1, b_hh1,
        c0 + BH, out_h1a, out_h1b, out_c1);
}